// GCNet_70059506532443
// MI455X (gfx1250) — compile-verified
//
#include <hip/hip_runtime.h>
#include <math.h>

typedef float v2f __attribute__((ext_vector_type(2)));
typedef float v8f __attribute__((ext_vector_type(8)));

#define NNODES   100000
#define NEDGES   3200000
#define NGRAPHS  512
#define DIN      128
#define NCLS     10
#define BN_EPS   1e-5f
#define NTILES   (NNODES / 16)   // 6250, exact

static __device__ __forceinline__ v8f wmma_f32(v2f a, v2f b, v8f c) {
    // V_WMMA_F32_16X16X4_F32 : D = A(16x4,f32) * B(4x16,f32) + C(16x16,f32)
    return __builtin_amdgcn_wmma_f32_16x16x4_f32(false, a, false, b, (short)0, c, false, false);
}

static __device__ __forceinline__ unsigned f2ord(float f) {
    unsigned u = __float_as_uint(f);
    return (u & 0x80000000u) ? ~u : (u | 0x80000000u);
}
static __device__ __forceinline__ float ord2f(unsigned u) {
    return (u & 0x80000000u) ? __uint_as_float(u & 0x7fffffffu) : __uint_as_float(~u);
}

// ---------------- init: deg=1 (self loop), pooled=ordered(-inf)=0, col stats=0 ----
__global__ void k_init(float* deg, unsigned* pooledU, float* colsum, float* colsq) {
    int i = blockIdx.x * blockDim.x + threadIdx.x;
    if (i < NNODES) deg[i] = 1.0f;
    if (i < NGRAPHS * 96) pooledU[i] = 0u;
    if (i < 96) { colsum[i] = 0.0f; colsq[i] = 0.0f; }
}

__global__ void k_degcount(const int* __restrict__ dst, float* deg) {
    int e = blockIdx.x * blockDim.x + threadIdx.x;
    if (e >= NEDGES) return;
    atomicAdd(deg + dst[e], 1.0f);
}

__global__ void k_dinv(float* deg) {
    int i = blockIdx.x * blockDim.x + threadIdx.x;
    if (i < NNODES) deg[i] = rsqrtf(deg[i]);   // in-place -> dinv
}

// ---------------- GEMM1: H1[100000,32] = X[100000,128] @ W1[128,32] (WMMA f32) ----
// B is staged in LDS pair-interleaved: Wp[(k/2)*2N + 2c + (k&1)] = W[k][c], so each
// lane's B operand (K-pair for one column) is a single aligned ds_load_b64.
__global__ void k_gemm1(const float* __restrict__ X, const float* __restrict__ W,
                        float* __restrict__ H1) {
    __shared__ float Ws[DIN * 32];
    for (int i = threadIdx.x; i < DIN * 32; i += blockDim.x) {
        int k = i >> 5, c = i & 31;
        Ws[(k >> 1) * 64 + c * 2 + (k & 1)] = W[i];
    }
    __syncthreads();
    int lane  = threadIdx.x & 31;
    int laneM = lane & 15;
    int hi    = lane >> 4;
    int tile  = blockIdx.x * (blockDim.x >> 5) + (threadIdx.x >> 5);
    if (tile >= NTILES) return;
    int m0 = tile * 16;
    const float* arow = X + (size_t)(m0 + laneM) * DIN;
    v8f acc0 = {}; v8f acc1 = {};
    for (int k = 0; k < DIN; k += 4) {
        v2f a = *(const v2f*)(arow + k + 2 * hi);
        int rp = ((k >> 1) + hi) * 64;               // pair-row base (kb = k + 2*hi, even)
        v2f b0 = *(const v2f*)(Ws + rp + 2 * laneM);
        v2f b1 = *(const v2f*)(Ws + rp + 2 * (16 + laneM));
        acc0 = wmma_f32(a, b0, acc0);
        acc1 = wmma_f32(a, b1, acc1);
    }
    float* orow = H1 + (size_t)(m0 + 8 * hi) * 32 + laneM;
    #pragma unroll
    for (int v = 0; v < 8; ++v) {
        orow[(size_t)v * 32]      = acc0[v];
        orow[(size_t)v * 32 + 16] = acc1[v];
    }
}

// -------- self-loop + bias init of destination buffer (layer1 -> hcat cols 0..31)
__global__ void k_selfinit1(const float* __restrict__ H1, const float* __restrict__ dinv,
                            const float* __restrict__ b1, float* __restrict__ Hc) {
    int i = blockIdx.x * blockDim.x + threadIdx.x;
    if (i >= NNODES * 32) return;
    int n = i >> 5, c = i & 31;
    float di = dinv[n];
    Hc[(size_t)n * 96 + c] = H1[i] * di * di + b1[c];
}

__global__ void k_scatter1(const int* __restrict__ src, const int* __restrict__ dst,
                           const float* __restrict__ dinv, const float* __restrict__ H1,
                           float* __restrict__ Hc) {
    int e = blockIdx.x * blockDim.x + threadIdx.x;
    if (e >= NEDGES) return;
    int s = src[e], d = dst[e];
    float nrm = dinv[s] * dinv[d];
    const float* hs = H1 + (size_t)s * 32;
    float* od = Hc + (size_t)d * 96;
    #pragma unroll
    for (int c = 0; c < 32; c += 4) {
        float4 v = *(const float4*)(hs + c);
        atomicAdd(od + c + 0, v.x * nrm);
        atomicAdd(od + c + 1, v.y * nrm);
        atomicAdd(od + c + 2, v.z * nrm);
        atomicAdd(od + c + 3, v.w * nrm);
    }
}

// ---------------- GEMM2: H2[100000,64] = x1(hcat cols0..31, ld=96) @ W2[32,64] ----
__global__ void k_gemm2(const float* __restrict__ Hc, const float* __restrict__ W,
                        float* __restrict__ H2) {
    __shared__ float Ws[32 * 64];
    for (int i = threadIdx.x; i < 32 * 64; i += blockDim.x) {
        int k = i >> 6, c = i & 63;
        Ws[(k >> 1) * 128 + c * 2 + (k & 1)] = W[i];
    }
    __syncthreads();
    int lane  = threadIdx.x & 31;
    int laneM = lane & 15;
    int hi    = lane >> 4;
    int tile  = blockIdx.x * (blockDim.x >> 5) + (threadIdx.x >> 5);
    if (tile >= NTILES) return;
    int m0 = tile * 16;
    const float* arow = Hc + (size_t)(m0 + laneM) * 96;   // cols 0..31 hold x1
    v8f acc[4] = {};
    for (int k = 0; k < 32; k += 4) {
        v2f a = *(const v2f*)(arow + k + 2 * hi);
        int rp = ((k >> 1) + hi) * 128;
        #pragma unroll
        for (int t = 0; t < 4; ++t) {
            v2f b = *(const v2f*)(Ws + rp + 2 * (t * 16 + laneM));
            acc[t] = wmma_f32(a, b, acc[t]);
        }
    }
    float* orow = H2 + (size_t)(m0 + 8 * hi) * 64 + laneM;
    #pragma unroll
    for (int t = 0; t < 4; ++t)
        #pragma unroll
        for (int v = 0; v < 8; ++v)
            orow[(size_t)v * 64 + t * 16] = acc[t][v];
}

__global__ void k_selfinit2(const float* __restrict__ H2, const float* __restrict__ dinv,
                            const float* __restrict__ b2, float* __restrict__ Hc) {
    int i = blockIdx.x * blockDim.x + threadIdx.x;
    if (i >= NNODES * 64) return;
    int n = i >> 6, c = i & 63;
    float di = dinv[n];
    Hc[(size_t)n * 96 + 32 + c] = H2[i] * di * di + b2[c];
}

__global__ void k_scatter2(const int* __restrict__ src, const int* __restrict__ dst,
                           const float* __restrict__ dinv, const float* __restrict__ H2,
                           float* __restrict__ Hc) {
    int e = blockIdx.x * blockDim.x + threadIdx.x;
    if (e >= NEDGES) return;
    int s = src[e], d = dst[e];
    float nrm = dinv[s] * dinv[d];
    const float* hs = H2 + (size_t)s * 64;
    float* od = Hc + (size_t)d * 96 + 32;
    #pragma unroll
    for (int c = 0; c < 64; c += 4) {
        float4 v = *(const float4*)(hs + c);
        atomicAdd(od + c + 0, v.x * nrm);
        atomicAdd(od + c + 1, v.y * nrm);
        atomicAdd(od + c + 2, v.z * nrm);
        atomicAdd(od + c + 3, v.w * nrm);
    }
}

// --- GEMM3 fused: Hl = relu(Hc[100000,96] @ lin1_W[96,96] + b), + BN column stats
__global__ void k_gemm3(const float* __restrict__ Hc, const float* __restrict__ W,
                        const float* __restrict__ bias, float* __restrict__ Hl,
                        float* __restrict__ colsum, float* __restrict__ colsq) {
    __shared__ float Ws[96 * 96];
    __shared__ float sb[96];
    __shared__ float ssum[96], ssq[96];
    for (int i = threadIdx.x; i < 96 * 96; i += blockDim.x) {
        int k = i / 96, c = i - k * 96;
        Ws[(k >> 1) * 192 + c * 2 + (k & 1)] = W[i];
    }
    for (int i = threadIdx.x; i < 96; i += blockDim.x) { sb[i] = bias[i]; ssum[i] = 0.f; ssq[i] = 0.f; }
    __syncthreads();
    int lane  = threadIdx.x & 31;
    int laneM = lane & 15;
    int hi    = lane >> 4;
    int tile  = blockIdx.x * (blockDim.x >> 5) + (threadIdx.x >> 5);
    bool active = (tile < NTILES);
    if (active) {
        int m0 = tile * 16;
        const float* arow = Hc + (size_t)(m0 + laneM) * 96;
        v8f acc[6] = {};
        for (int k = 0; k < 96; k += 4) {
            v2f a = *(const v2f*)(arow + k + 2 * hi);
            int rp = ((k >> 1) + hi) * 192;
            #pragma unroll
            for (int t = 0; t < 6; ++t) {
                v2f b = *(const v2f*)(Ws + rp + 2 * (t * 16 + laneM));
                acc[t] = wmma_f32(a, b, acc[t]);
            }
        }
        #pragma unroll
        for (int t = 0; t < 6; ++t) {
            int col = t * 16 + laneM;
            float bb = sb[col];
            float psum = 0.f, psq = 0.f;
            #pragma unroll
            for (int v = 0; v < 8; ++v) {
                float val = fmaxf(acc[t][v] + bb, 0.0f);
                Hl[(size_t)(m0 + v + 8 * hi) * 96 + col] = val;
                psum += val; psq += val * val;
            }
            atomicAdd(&ssum[col], psum);
            atomicAdd(&ssq[col], psq);
        }
    }
    __syncthreads();
    for (int i = threadIdx.x; i < 96; i += blockDim.x) {
        atomicAdd(&colsum[i], ssum[i]);
        atomicAdd(&colsq[i], ssq[i]);
    }
}

// ---------------- BN stats -> affine scale/shift -----------------------------
__global__ void k_stats(const float* colsum, const float* colsq,
                        const float* g, const float* bt, float* scale, float* shift) {
    int c = threadIdx.x;
    if (c >= 96) return;
    float m   = colsum[c] / (float)NNODES;
    float var = colsq[c] / (float)NNODES - m * m;
    float istd = rsqrtf(var + BN_EPS);
    float sc = g[c] * istd;
    scale[c] = sc;
    shift[c] = bt[c] - m * sc;
}

// ---------------- BN apply + segment_max pool (ordered-u32 atomicMax) ---------
__global__ void k_normpool(const float* __restrict__ Hl, const float* __restrict__ scale,
                           const float* __restrict__ shift, const int* __restrict__ batch,
                           unsigned* __restrict__ pooledU) {
    int i = blockIdx.x * blockDim.x + threadIdx.x;
    if (i >= NNODES * 96) return;
    int n = i / 96, c = i - n * 96;
    float val = Hl[i] * scale[c] + shift[c];
    atomicMax(pooledU + (size_t)batch[n] * 96 + c, f2ord(val));
}

// ---------------- head: decode pool -> MLP(96) -> BN -> MLP(10) -> BN ---------
__global__ void __launch_bounds__(1024)
k_head(const unsigned* __restrict__ pooledU, const float* __restrict__ m1W,
       const float* __restrict__ m1b, const float* __restrict__ m1g, const float* __restrict__ m1bt,
       const float* __restrict__ m2W, const float* __restrict__ m2b,
       const float* __restrict__ m2g, const float* __restrict__ m2bt,
       float* __restrict__ pooledF, float* __restrict__ H4, float* __restrict__ out) {
    __shared__ float sW1[96 * 96];   // 36 KB
    __shared__ float sW2[96 * NCLS]; // ~3.8 KB
    __shared__ float s5[NGRAPHS * NCLS]; // 20 KB
    __shared__ float sscale[96], sshift[96];
    __shared__ float s5m[NCLS], s5i[NCLS];
    int tid = threadIdx.x, bd = blockDim.x;
    for (int i = tid; i < 96 * 96; i += bd) sW1[i] = m1W[i];
    for (int i = tid; i < 96 * NCLS; i += bd) sW2[i] = m2W[i];
    for (int i = tid; i < NGRAPHS * 96; i += bd) {
        float f = ord2f(pooledU[i]);
        if (!isfinite(f)) f = 0.0f;      // empty-segment guard, matches reference
        pooledF[i] = f;
    }
    __threadfence_block(); __syncthreads();
    // h4 = relu(pooledF @ m1W + m1b)
    for (int o = tid; o < NGRAPHS * 96; o += bd) {
        int r = o / 96, c = o - r * 96;
        const float* pr = pooledF + (size_t)r * 96;
        float a = m1b[c];
        for (int k = 0; k < 96; ++k) a += pr[k] * sW1[k * 96 + c];
        H4[o] = fmaxf(a, 0.0f);
    }
    __threadfence_block(); __syncthreads();
    if (tid < 96) {
        float s = 0.f, q = 0.f;
        for (int r = 0; r < NGRAPHS; ++r) { float v = H4[r * 96 + tid]; s += v; q += v * v; }
        float m = s / (float)NGRAPHS;
        float var = q / (float)NGRAPHS - m * m;
        float istd = rsqrtf(var + BN_EPS);
        sscale[tid] = m1g[tid] * istd;
        sshift[tid] = m1bt[tid] - m * m1g[tid] * istd;
    }
    __syncthreads();
    // h5 = relu(bn(h4) @ m2W + m2b)
    for (int o = tid; o < NGRAPHS * NCLS; o += bd) {
        int r = o / NCLS, c = o - r * NCLS;
        const float* hr = H4 + (size_t)r * 96;
        float a = m2b[c];
        for (int k = 0; k < 96; ++k) a += (hr[k] * sscale[k] + sshift[k]) * sW2[k * NCLS + c];
        s5[o] = fmaxf(a, 0.0f);
    }
    __syncthreads();
    if (tid < NCLS) {
        float s = 0.f, q = 0.f;
        for (int r = 0; r < NGRAPHS; ++r) { float v = s5[r * NCLS + tid]; s += v; q += v * v; }
        float m = s / (float)NGRAPHS;
        float var = q / (float)NGRAPHS - m * m;
        s5m[tid] = m;
        s5i[tid] = m2g[tid] * rsqrtf(var + BN_EPS);
    }
    __syncthreads();
    for (int o = tid; o < NGRAPHS * NCLS; o += bd) {
        int c = o % NCLS;
        out[o] = s5i[c] * (s5[o] - s5m[c]) + m2bt[c];
    }
}

extern "C" void kernel_launch(void* const* d_in, const int* in_sizes, int n_in,
                              void* d_out, int out_size, void* d_ws, size_t ws_size,
                              hipStream_t stream) {
    const float* x      = (const float*)d_in[0];
    const int*   eidx   = (const int*)d_in[1];
    const int*   batch  = (const int*)d_in[2];
    const float* W1     = (const float*)d_in[3];
    const float* b1     = (const float*)d_in[4];
    const float* W2     = (const float*)d_in[5];
    const float* b2     = (const float*)d_in[6];
    const float* lin1W  = (const float*)d_in[7];
    const float* lin1b  = (const float*)d_in[8];
    const float* lin1g  = (const float*)d_in[9];
    const float* lin1bt = (const float*)d_in[10];
    const float* m1W    = (const float*)d_in[11];
    const float* m1b    = (const float*)d_in[12];
    const float* m1g    = (const float*)d_in[13];
    const float* m1bt   = (const float*)d_in[14];
    const float* m2W    = (const float*)d_in[15];
    const float* m2b    = (const float*)d_in[16];
    const float* m2g    = (const float*)d_in[17];
    const float* m2bt   = (const float*)d_in[18];
    float* outp = (float*)d_out;

    const int* src = eidx;
    const int* dst = eidx + NEDGES;

    float* ws = (float*)d_ws;
    float*    deg     = ws;                        // 100352 (-> dinv in place)
    float*    h1      = deg + 100352;              // 100000*32
    float*    hcat    = h1 + (size_t)NNODES * 32;  // 100000*96
    float*    h2      = hcat + (size_t)NNODES * 96;// 100000*64
    float*    hlin    = h2 + (size_t)NNODES * 64;  // 100000*96
    float*    colsum  = hlin + (size_t)NNODES * 96;// 128
    float*    colsq   = colsum + 128;              // 128
    float*    scale   = colsq + 128;               // 128
    float*    shift   = scale + 128;               // 128
    unsigned* pooledU = (unsigned*)(shift + 128);  // 512*96
    float*    pooledF = (float*)(pooledU + NGRAPHS * 96);
    float*    h4      = pooledF + NGRAPHS * 96;    // 512*96

    const int T = 256;
    const int gemm_blocks = (NTILES + (T / 32) - 1) / (T / 32); // 6250 tiles, 8 waves/blk

    k_init<<<(NNODES + T - 1) / T, T, 0, stream>>>(deg, pooledU, colsum, colsq);
    k_degcount<<<(NEDGES + T - 1) / T, T, 0, stream>>>(dst, deg);
    k_dinv<<<(NNODES + T - 1) / T, T, 0, stream>>>(deg);

    k_gemm1<<<gemm_blocks, T, 0, stream>>>(x, W1, h1);
    k_selfinit1<<<(NNODES * 32 + T - 1) / T, T, 0, stream>>>(h1, deg, b1, hcat);
    k_scatter1<<<(NEDGES + T - 1) / T, T, 0, stream>>>(src, dst, deg, h1, hcat);

    k_gemm2<<<gemm_blocks, T, 0, stream>>>(hcat, W2, h2);
    k_selfinit2<<<(NNODES * 64 + T - 1) / T, T, 0, stream>>>(h2, deg, b2, hcat);
    k_scatter2<<<(NEDGES + T - 1) / T, T, 0, stream>>>(src, dst, deg, h2, hcat);

    k_gemm3<<<gemm_blocks, T, 0, stream>>>(hcat, lin1W, lin1b, hlin, colsum, colsq);
    k_stats<<<1, 128, 0, stream>>>(colsum, colsq, lin1g, lin1bt, scale, shift);
    k_normpool<<<(NNODES * 96 + T - 1) / T, T, 0, stream>>>(hlin, scale, shift, batch, pooledU);

    k_head<<<1, 1024, 0, stream>>>(pooledU, m1W, m1b, m1g, m1bt,
                                   m2W, m2b, m2g, m2bt, pooledF, h4, outp);
}